// SimpleSortNet_22265110463113
// MI455X (gfx1250) — compile-verified
//
#include <hip/hip_runtime.h>
#include <hip/hip_bf16.h>

#define HEADS 8
#define NB 128           // MAX_BUCKETS
#define DIM 128
#define DH 64
#define T 16384
#define B 4
#define BH (B * HEADS)   // 32
#define EPSV 1e-6f
#define INV_TEMP (1.0f / 0.7f)

typedef __attribute__((ext_vector_type(2))) float v2f;
typedef __attribute__((ext_vector_type(8))) float v8f;

// ---------------------------------------------------------------------------
// Kernel 0: zero the x accumulator (bh, NB, DIM) = 32*128*128 floats in ws
// ---------------------------------------------------------------------------
__global__ __launch_bounds__(256) void k_zero(float* __restrict__ x, int n) {
    int i = blockIdx.x * 256 + threadIdx.x;
    if (i < n) x[i] = 0.0f;
}

// ---------------------------------------------------------------------------
// Kernel 1: segmented sum of q,k into x[bh][bucket][0:64]=sum q, [64:128]=sum k
// ids are sorted per sample -> run-length accumulate in registers, flush with
// one global f32 atomic per (bucket-change, dim).
// grid: (BH, T/256), block: 256 = 4 row-groups x 64 dims
// ---------------------------------------------------------------------------
__global__ __launch_bounds__(256) void k_segsum(const float* __restrict__ q,
                                                const float* __restrict__ k,
                                                const int* __restrict__ seg,
                                                float* __restrict__ x) {
    const int bh    = blockIdx.x;           // 0..31
    const int chunk = blockIdx.y;           // 0..63
    const int b     = bh >> 3;              // sample
    const int d     = threadIdx.x & 63;     // dim 0..63
    const int rg    = threadIdx.x >> 6;     // row group 0..3
    const int t0    = chunk * 256 + rg * 64;

    const float* qp = q + ((size_t)bh * T + t0) * DH + d;
    const float* kp = k + ((size_t)bh * T + t0) * DH + d;
    const int*   sp = seg + (size_t)b * T + t0;
    float*       xb = x + (size_t)bh * NB * DIM;

    float accq = 0.0f, acck = 0.0f;
    int cur = sp[0];
#pragma unroll 4
    for (int i = 0; i < 64; ++i) {
        int s = sp[i];
        if (s != cur) {
            atomicAdd(&xb[cur * DIM + d],      accq);
            atomicAdd(&xb[cur * DIM + DH + d], acck);
            accq = 0.0f; acck = 0.0f; cur = s;
        }
        accq += qp[(size_t)i * DH];
        acck += kp[(size_t)i * DH];
    }
    atomicAdd(&xb[cur * DIM + d],      accq);
    atomicAdd(&xb[cur * DIM + DH + d], acck);
}

// ---------------------------------------------------------------------------
// Kernel 2: R = relu(x @ W[h]) via V_WMMA_F32_16X16X4_F32, fused with
// gumbel-log epilogue; writes pre-sinkhorn r into out (bh, NB, NB).
// grid: (BH, 8 tile-rows), block 256 = 8 waves; wave w computes tile col w.
// ---------------------------------------------------------------------------
__global__ __launch_bounds__(256) void k_gemm(const float* __restrict__ x,
                                              const float* __restrict__ linear,
                                              const float* __restrict__ gumbel,
                                              float* __restrict__ out) {
    const int bh    = blockIdx.x;           // 0..31
    const int trow  = blockIdx.y;           // 0..7
    const int h     = bh & 7;
    const int wave  = threadIdx.x >> 5;     // tile col 0..7
    const int lane  = threadIdx.x & 31;
    const int lhalf = lane >> 4;            // 0|1
    const int l16   = lane & 15;

    // A row for this lane (matrix row M = trow*16 + l16), row-major K stride 1
    const float* A = x + (size_t)bh * NB * DIM + (trow * 16 + l16) * DIM;
    // B column for this lane (matrix col N = wave*16 + l16), row-major K stride DIM... -> stride NB
    const float* Bp = linear + (size_t)h * DIM * NB + (wave * 16 + l16);

    v8f acc = {};
#if defined(__has_builtin)
#if __has_builtin(__builtin_amdgcn_wmma_f32_16x16x4_f32)
#define HAVE_WMMA_F32 1
#endif
#endif
#ifdef HAVE_WMMA_F32
    for (int kk = 0; kk < DIM; kk += 4) {
        const int ka = kk + 2 * lhalf;           // lanes 0-15: K=kk,kk+1 ; 16-31: K=kk+2,kk+3
        v2f a, bb;
        a.x  = A[ka];
        a.y  = A[ka + 1];
        bb.x = Bp[(size_t)ka * NB];
        bb.y = Bp[(size_t)(ka + 1) * NB];
        acc = __builtin_amdgcn_wmma_f32_16x16x4_f32(
            /*neg_a=*/false, a, /*neg_b=*/false, bb,
            /*c_mod=*/(short)0, acc, /*reuse_a=*/false, /*reuse_b=*/false);
    }
#else
    // scalar fallback matching the C/D VGPR layout (M = r + 8*lhalf, N = l16)
    for (int r = 0; r < 8; ++r) {
        const int M = trow * 16 + r + 8 * lhalf;
        const float* Arow = x + (size_t)bh * NB * DIM + M * DIM;
        float s = 0.0f;
        for (int kk = 0; kk < DIM; ++kk) s = fmaf(Arow[kk], Bp[(size_t)kk * NB], s);
        acc[r] = s;
    }
#endif

    // epilogue: D element r -> (M = trow*16 + r + 8*lhalf, N = wave*16 + l16)
    const int mcol = wave * 16 + l16;
    const float* G = gumbel + (size_t)bh * NB * NB;
    float*       O = out    + (size_t)bh * NB * NB;
#pragma unroll
    for (int r = 0; r < 8; ++r) {
        const int n = trow * 16 + r + 8 * lhalf;
        float v  = fmaxf(acc[r], 0.0f);
        float rr = __logf(v + EPSV);
        float u  = G[n * NB + mcol];
        float g  = -__logf(-__logf(u + EPSV) + EPSV);
        O[n * NB + mcol] = (rr + g) * INV_TEMP;
    }
}

// ---------------------------------------------------------------------------
// Kernel 3: 8 Sinkhorn iterations (row-lse then col-lse) + exp, in-place on out.
// one block per bh; 128x128 f32 tile in LDS (64 KB).
// ---------------------------------------------------------------------------
__global__ __launch_bounds__(256) void k_sinkhorn(float* __restrict__ out) {
    __shared__ float tile[NB * NB];
    const int bh  = blockIdx.x;
    const int tid = threadIdx.x;
    float* base = out + (size_t)bh * NB * NB;

    for (int i = tid; i < NB * NB; i += 256) tile[i] = base[i];
    __syncthreads();

    for (int it = 0; it < 8; ++it) {
        // logsumexp over axis 2 (columns of a row)
        if (tid < NB) {
            const int n = tid;
            float m = -3.402823466e+38f;
            for (int j = 0; j < NB; ++j) m = fmaxf(m, tile[n * NB + j]);
            float s = 0.0f;
            for (int j = 0; j < NB; ++j) s += __expf(tile[n * NB + j] - m);
            const float lse = m + __logf(s);
            for (int j = 0; j < NB; ++j) tile[n * NB + j] -= lse;
        }
        __syncthreads();
        // logsumexp over axis 1 (rows of a column)
        if (tid < NB) {
            const int j = tid;
            float m = -3.402823466e+38f;
            for (int n = 0; n < NB; ++n) m = fmaxf(m, tile[n * NB + j]);
            float s = 0.0f;
            for (int n = 0; n < NB; ++n) s += __expf(tile[n * NB + j] - m);
            const float lse = m + __logf(s);
            for (int n = 0; n < NB; ++n) tile[n * NB + j] -= lse;
        }
        __syncthreads();
    }

    for (int i = tid; i < NB * NB; i += 256) base[i] = __expf(tile[i]);
}

// ---------------------------------------------------------------------------
extern "C" void kernel_launch(void* const* d_in, const int* in_sizes, int n_in,
                              void* d_out, int out_size, void* d_ws, size_t ws_size,
                              hipStream_t stream) {
    const float* q      = (const float*)d_in[0];   // (32, 16384, 64)
    const float* k      = (const float*)d_in[1];   // (32, 16384, 64)
    const float* linear = (const float*)d_in[2];   // (1, 8, 128, 128)
    const int*   seg    = (const int*)d_in[3];     // (4, 16384)
    const float* gum    = (const float*)d_in[4];   // (32, 128, 128)
    float* out = (float*)d_out;                    // (32, 128, 128)
    float* x   = (float*)d_ws;                     // (32, 128, 128) accumulator

    const int xcount = BH * NB * DIM;              // 524288

    k_zero<<<dim3((xcount + 255) / 256), dim3(256), 0, stream>>>(x, xcount);
    k_segsum<<<dim3(BH, T / 256), dim3(256), 0, stream>>>(q, k, seg, x);
    k_gemm<<<dim3(BH, NB / 16), dim3(256), 0, stream>>>(x, linear, gum, out);
    k_sinkhorn<<<dim3(BH), dim3(256), 0, stream>>>(out);
}